// LinOSS_27066883899580
// MI455X (gfx1250) — compile-verified
//
#include <hip/hip_runtime.h>

#define BDIM 8
#define TDIM 2048
#define HDIM 16
#define DDIM 64
#define DT_C 0.01f
#define CHUNK 16
#define NCHUNK (TDIM / CHUNK)
#define COLSPLIT 2                 // blocks per (b,h): each owns 64/COLSPLIT columns

typedef __attribute__((ext_vector_type(2))) float v2f;

// ---- CDNA5 packed fp32 VOP3P ops (ISA §15.10: V_PK_MUL_F32 / V_PK_FMA_F32) ----
__device__ __forceinline__ v2f pk_mul(v2f a, v2f b) {
  v2f d;
  asm("v_pk_mul_f32 %0, %1, %2" : "=v"(d) : "v"(a), "v"(b));
  return d;
}
__device__ __forceinline__ v2f pk_fma(v2f a, v2f b, v2f c) {
  v2f d;
  asm("v_pk_fma_f32 %0, %1, %2, %3" : "=v"(d) : "v"(a), "v"(b), "v"(c));
  return d;
}

// ---- CDNA5 async global->LDS helpers (ISA: GLOBAL_LOAD_ASYNC_TO_LDS_*) ----
__device__ __forceinline__ unsigned lds_off(const void* p) {
  return (unsigned)(uintptr_t)p;  // low 32 bits of generic LDS ptr = LDS byte offset
}
__device__ __forceinline__ void async_ld_b128(unsigned lds, const void* gaddr) {
  asm volatile("global_load_async_to_lds_b128 %0, %1, off"
               :: "v"(lds), "v"(gaddr) : "memory");
}
__device__ __forceinline__ void async_ld_b32(unsigned lds, const void* gaddr) {
  asm volatile("global_load_async_to_lds_b32 %0, %1, off"
               :: "v"(lds), "v"(gaddr) : "memory");
}

__device__ __forceinline__ void issue_chunk(
    const float* __restrict__ gk, const float* __restrict__ gv,
    const float* __restrict__ gq, const float* __restrict__ gb,
    size_t row_base, int t0, int r, int sg,
    float* sk, float* sv, float* sq, float* sb)
{
  const size_t row = row_base + (size_t)(t0 + r) * HDIM;        // (b*T + t)*H + h
  const size_t go  = (row * DDIM + (size_t)sg * 4) * sizeof(float);
  async_ld_b128(lds_off(sk + r * DDIM + sg * 4), (const char*)gk + go);
  async_ld_b128(lds_off(sv + r * DDIM + sg * 4), (const char*)gv + go);
  async_ld_b128(lds_off(sq + r * DDIM + sg * 4), (const char*)gq + go);
  const size_t gob = (row_base + (size_t)(t0 + sg) * HDIM) * sizeof(float);
  async_ld_b32(lds_off(sb + sg), (const char*)gb + gob);   // 16x dup, same value
}

extern "C" __global__ __launch_bounds__(256)
void linoss_scan(const float* __restrict__ q, const float* __restrict__ k,
                 const float* __restrict__ v, const float* __restrict__ beta,
                 const float* __restrict__ osc_w, const float* __restrict__ osc_damp,
                 float* __restrict__ out)
{
  __shared__ __align__(16) float s_k[2][CHUNK * DDIM];
  __shared__ __align__(16) float s_v[2][CHUNK * DDIM];
  __shared__ __align__(16) float s_q[2][CHUNK * DDIM];
  __shared__ __align__(16) float s_b[2][CHUNK];

  const int tid  = threadIdx.x;
  const int w    = tid >> 5;                 // wave 0..7
  const int l    = tid & 31;                 // lane
  const int bh   = blockIdx.x >> 1;          // (b,h) pair (COLSPLIT == 2)
  const int half = blockIdx.x & 1;           // which 32-column half
  const int b    = bh >> 4;                  // H == 16
  const int h    = bh & 15;

  // lane owns 8 rows x 1 column:
  const int e  = half * 32 + (w << 2) + (l & 3);   // column 0..63
  const int d0 = (l >> 2) << 3;                    // rows d0..d0+7

  // ---- per-lane coefficients + state as 4 packed f32 pairs ----
  // z' = c1*z + (DT*beta)*(k_d v_e) + c2n*y ; c1 = 1 - sigmoid(osc_damp),
  //                                           c2n = -DT*exp(osc_w)
  // y' = y + DT*z'
  v2f c1p[4], c2n[4], y2[4], z2[4];
  {
    const size_t obase = (size_t)h * DDIM * DDIM + (size_t)d0 * DDIM + e;
#pragma unroll
    for (int i = 0; i < 4; ++i) {
      const float ow0 = osc_w[obase + (size_t)(2 * i)     * DDIM];
      const float ow1 = osc_w[obase + (size_t)(2 * i + 1) * DDIM];
      const float od0 = osc_damp[obase + (size_t)(2 * i)     * DDIM];
      const float od1 = osc_damp[obase + (size_t)(2 * i + 1) * DDIM];
      c2n[i] = (v2f){-DT_C * __expf(ow0), -DT_C * __expf(ow1)};
      c1p[i] = (v2f){1.0f - 1.0f / (1.0f + __expf(-od0)),
                     1.0f - 1.0f / (1.0f + __expf(-od1))};
      y2[i] = (v2f){0.0f, 0.0f};
      z2[i] = (v2f){0.0f, 0.0f};
    }
  }
  const v2f dt2 = (v2f){DT_C, DT_C};

  const int    r        = tid >> 4;   // staging: chunk-row 0..15
  const int    sg       = tid & 15;   // staging: 16B segment
  const size_t row_base = (size_t)b * TDIM * HDIM + h;   // (b*T)*H + h

  issue_chunk(k, v, q, beta, row_base, 0, r, sg,
              s_k[0], s_v[0], s_q[0], s_b[0]);

  for (int c = 0; c < NCHUNK; ++c) {
    const int buf = c & 1;
    if (c + 1 < NCHUNK) {
      issue_chunk(k, v, q, beta, row_base, (c + 1) * CHUNK, r, sg,
                  s_k[buf ^ 1], s_v[buf ^ 1], s_q[buf ^ 1], s_b[buf ^ 1]);
      asm volatile("s_wait_asynccnt 0x4" ::: "memory");   // chunk c complete
    } else {
      asm volatile("s_wait_asynccnt 0x0" ::: "memory");
    }
    __syncthreads();

    const float*  sk  = s_k[buf];
    const float*  sv  = s_v[buf];
    const float*  sq  = s_q[buf];
    const float4* sb4 = (const float4*)s_b[buf];
    const int t0 = c * CHUNK;

    for (int u = 0; u < 4; ++u) {          // 4 groups of 4 steps
      const float4 bb = sb4[u];            // one LDS load per 4 steps
#pragma unroll
      for (int p = 0; p < 4; ++p) {
        const int   t  = u * 4 + p;
        const float bt = (p == 0) ? bb.x : (p == 1) ? bb.y : (p == 2) ? bb.z : bb.w;
        const float vvs = (DT_C * bt) * sv[t * DDIM + e];   // DT*beta*v_e
        const v2f vv2 = (v2f){vvs, vvs};
        const float4* kp = (const float4*)(sk + t * DDIM + d0);
        const float4* qp = (const float4*)(sq + t * DDIM + d0);
        v2f acc2 = (v2f){0.0f, 0.0f};
#pragma unroll
        for (int j = 0; j < 2; ++j) {      // 2 x float4 = rows d0..d0+7
          const float4 kk = kp[j];
          const float4 qq = qp[j];
          const v2f ka = (v2f){kk.x, kk.y}, kb = (v2f){kk.z, kk.w};
          const v2f qa = (v2f){qq.x, qq.y}, qb = (v2f){qq.z, qq.w};
          const int i = j * 2;
          z2[i]     = pk_fma(c1p[i],     z2[i],     pk_fma(ka, vv2, pk_mul(c2n[i],     y2[i])));
          y2[i]     = pk_fma(dt2,        z2[i],     y2[i]);
          acc2      = pk_fma(qa,         y2[i],     acc2);
          z2[i + 1] = pk_fma(c1p[i + 1], z2[i + 1], pk_fma(kb, vv2, pk_mul(c2n[i + 1], y2[i + 1])));
          y2[i + 1] = pk_fma(dt2,        z2[i + 1], y2[i + 1]);
          acc2      = pk_fma(qb,         y2[i + 1], acc2);
        }
        float acc = acc2.x + acc2.y;
        // reduce over the 8 row-groups (lanes l ^ {4,8,16}) — intra-wave
        acc += __shfl_xor(acc, 4, 32);
        acc += __shfl_xor(acc, 8, 32);
        acc += __shfl_xor(acc, 16, 32);
        if (l < 4) {   // lanes 0..3 hold columns half*32 + w*4 + l
          float* op = &out[(row_base + (size_t)(t0 + t) * HDIM) * DDIM +
                           half * 32 + (w << 2) + l];
          __builtin_nontemporal_store(acc * 0.125f, op);   // q scale = D^-0.5
        }
      }
    }
    __syncthreads();   // buffer free before chunk c+2 overwrites it
  }
}

extern "C" void kernel_launch(void* const* d_in, const int* in_sizes, int n_in,
                              void* d_out, int out_size, void* d_ws, size_t ws_size,
                              hipStream_t stream) {
  const float* q        = (const float*)d_in[0];
  const float* k        = (const float*)d_in[1];
  const float* v        = (const float*)d_in[2];
  const float* beta     = (const float*)d_in[3];
  const float* osc_w    = (const float*)d_in[4];
  const float* osc_damp = (const float*)d_in[5];
  float* out = (float*)d_out;
  (void)in_sizes; (void)n_in; (void)out_size; (void)d_ws; (void)ws_size;

  hipLaunchKernelGGL(linoss_scan, dim3(BDIM * HDIM * COLSPLIT), dim3(256), 0, stream,
                     q, k, v, beta, osc_w, osc_damp, out);
}